// PoseVelGraph_36928128811471
// MI455X (gfx1250) — compile-verified
//
#include <hip/hip_runtime.h>
#include <cstdint>
#include <cstddef>

typedef unsigned int v4u __attribute__((ext_vector_type(4)));
typedef int          v8i __attribute__((ext_vector_type(8)));
typedef int          v4i __attribute__((ext_vector_type(4)));

#define EPS_F 1e-8f

struct V3 { float x, y, z; };
struct Q4 { float x, y, z, w; };

__device__ __forceinline__ V3 v3add(V3 a, V3 b){ return {a.x+b.x, a.y+b.y, a.z+b.z}; }
__device__ __forceinline__ V3 v3sub(V3 a, V3 b){ return {a.x-b.x, a.y-b.y, a.z-b.z}; }
__device__ __forceinline__ V3 v3scale(float s, V3 a){ return {s*a.x, s*a.y, s*a.z}; }
__device__ __forceinline__ float v3dot(V3 a, V3 b){ return a.x*b.x + a.y*b.y + a.z*b.z; }
__device__ __forceinline__ V3 v3cross(V3 a, V3 b){
  return { a.y*b.z - a.z*b.y, a.z*b.x - a.x*b.z, a.x*b.y - a.y*b.x };
}
__device__ __forceinline__ Q4 qconj(Q4 q){ return {-q.x, -q.y, -q.z, q.w}; }
__device__ __forceinline__ Q4 qmul(Q4 a, Q4 b){
  V3 va{a.x,a.y,a.z}, vb{b.x,b.y,b.z};
  V3 c = v3cross(va, vb);
  return { a.w*b.x + b.w*a.x + c.x,
           a.w*b.y + b.w*a.y + c.y,
           a.w*b.z + b.w*a.z + c.z,
           a.w*b.w - v3dot(va, vb) };
}
__device__ __forceinline__ V3 qrot(Q4 q, V3 v){
  V3 qv{q.x,q.y,q.z};
  V3 t = v3scale(2.f, v3cross(qv, v));
  return v3add(v3add(v, v3scale(q.w, t)), v3cross(qv, t));
}
__device__ __forceinline__ V3 so3_log(Q4 q){
  float s = (q.w < 0.f) ? -1.f : 1.f;
  float vx = s*q.x, vy = s*q.y, vz = s*q.z, w = s*q.w;
  float n2 = vx*vx + vy*vy + vz*vz;
  float n  = sqrtf(fmaxf(n2, EPS_F*EPS_F));
  float angle = 2.f * atan2f(n, w);
  float w_safe = (fabsf(w) > EPS_F) ? w : 1.f;
  float scale = (n2 > EPS_F) ? (angle / n) : (2.f / w_safe);
  return { scale*vx, scale*vy, scale*vz };
}
__device__ __forceinline__ void se3_log(V3 t, Q4 q, V3* tau, V3* phi_out){
  V3 phi = so3_log(q);
  float th2 = v3dot(phi, phi);
  float th  = sqrtf(fmaxf(th2, EPS_F*EPS_F));
  float half = 0.5f * th;
  float sh = sinf(fmaxf(half, EPS_F));
  float coef_big = (1.f - half * cosf(half) / sh) / fmaxf(th2, EPS_F*EPS_F);
  float coef = (th2 > EPS_F) ? coef_big : (1.f/12.f);
  V3 ct = v3cross(phi, t);
  *tau = v3add(v3sub(t, v3scale(0.5f, ct)), v3scale(coef, v3cross(phi, ct)));
  *phi_out = phi;
}

// ---------------------------------------------------------------------------
// Edge kernel: one thread per edge. The contiguous `poses` stream for this
// block's 256 edges (1792 dwords) is staged into LDS by the Tensor Data Mover
// (one tensor_load_to_lds per block, issued by wave 0, tracked by TENSORcnt),
// while node gathers go through normal global_load_b32 (L2-resident).
// ---------------------------------------------------------------------------
__global__ __launch_bounds__(256) void pg_edge_kernel(
    const float* __restrict__ nodes,
    const float* __restrict__ poses,
    const int*   __restrict__ edges,
    float*       __restrict__ out,
    int E)
{
  __shared__ float lds_pose[256 * 7];
  const int blockStart = blockIdx.x * 256;

  if (threadIdx.x < 32) {
    // Build Tensor DMA Descriptor (D#) — all block-uniform (SGPR) values.
    unsigned long long gaddr =
        (unsigned long long)(const void*)(poses + (size_t)blockStart * 7);
    unsigned int ldsb = (unsigned int)(size_t)(void*)lds_pose;  // LDS byte addr
    unsigned int td0  = 7u * (unsigned int)(E - blockStart);    // elems avail (OOB -> 0)

    v4u g0;                       // D# group 0 (128b)
    g0[0] = 1u;                                                   // count=1, user mode
    g0[1] = ldsb;                                                 // lds_addr
    g0[2] = (unsigned int)gaddr;                                  // global_addr[31:0]
    g0[3] = (unsigned int)((gaddr >> 32) & 0x1FFFFFFull)          // global_addr[56:32]
          | (2u << 30);                                           // type=2 ("image")

    v8i g1;                       // D# group 1 (256b)
    g1[0] = (int)(2u << 16);                 // workgroup_mask=0, data_size=4B
    g1[1] = (int)((td0 & 0xFFFFu) << 16);    // tensor_dim0[15:0]
    g1[2] = (int)((td0 >> 16) | (1u << 16)); // tensor_dim0[31:16], tensor_dim1=1
    g1[3] = (int)(1792u << 16);              // tile_dim0 = 1792 dwords
    g1[4] = 0;                               // tile_dim1=0 (unused), tile_dim2=0
    g1[5] = (int)td0;                        // tensor_dim0_stride[31:0]
    g1[6] = 0;                               // stride high / dim1 stride
    g1[7] = 0;

    v4i g2 = {0, 0, 0, 0};
    v4i g3 = {0, 0, 0, 0};

#if defined(__clang_major__) && (__clang_major__ >= 23)
    v8i g4 = {0, 0, 0, 0, 0, 0, 0, 0};
    __builtin_amdgcn_tensor_load_to_lds(g0, g1, g2, g3, g4, 0);
#else
    __builtin_amdgcn_tensor_load_to_lds(g0, g1, g2, g3, 0);
#endif
    __builtin_amdgcn_s_wait_tensorcnt(0);
  }
  __syncthreads();

  const int e = blockStart + (int)threadIdx.x;
  if (e >= E) return;

  const int ia = edges[2 * e + 0];
  const int ib = edges[2 * e + 1];
  const float* na = nodes + 7 * (size_t)ia;
  const float* nb = nodes + 7 * (size_t)ib;
  V3 t1{na[0], na[1], na[2]}; Q4 q1{na[3], na[4], na[5], na[6]};
  V3 t2{nb[0], nb[1], nb[2]}; Q4 q2{nb[3], nb[4], nb[5], nb[6]};

  const float* pp = &lds_pose[threadIdx.x * 7];
  V3 pt_in{pp[0], pp[1], pp[2]}; Q4 pq_in{pp[3], pp[4], pp[5], pp[6]};

  // (it1, iq1) = se3_inv(t1, q1); (mt, mq) = se3_mul(it1, iq1, t2, q2)
  Q4 iq1 = qconj(q1);
  V3 it1 = v3scale(-1.f, qrot(iq1, t1));
  V3 mt  = v3add(it1, qrot(iq1, t2));
  Q4 mq  = qmul(iq1, q2);

  // (pt, pq) = se3_inv(pose); (et, eq) = se3_mul(pt, pq, mt, mq)
  Q4 pq = qconj(pq_in);
  V3 pt = v3scale(-1.f, qrot(pq, pt_in));
  V3 et = v3add(pt, qrot(pq, mt));
  Q4 eq = qmul(pq, mq);

  V3 tau, phi;
  se3_log(et, eq, &tau, &phi);

  float* o = out + 6 * (size_t)e;           // L1 = 1.0
  o[0] = tau.x; o[1] = tau.y; o[2] = tau.z;
  o[3] = phi.x; o[4] = phi.y; o[5] = phi.z;
}

// ---------------------------------------------------------------------------
// Adjacent-node kernel: one thread per i in [0, N-1); also emits the 9 stop
// velocity residuals from block 0.
// ---------------------------------------------------------------------------
__global__ __launch_bounds__(256) void pg_adj_kernel(
    const float* __restrict__ nodes,
    const float* __restrict__ vels,
    const float* __restrict__ imu_drots,
    const float* __restrict__ imu_dvels,
    const float* __restrict__ imu_dtrans,
    const float* __restrict__ dts,
    float* __restrict__ out_adjvel,
    float* __restrict__ out_imurot,
    float* __restrict__ out_transvel,
    float* __restrict__ out_stop,
    int M)
{
  const int i = blockIdx.x * 256 + (int)threadIdx.x;
  if (i < M) {
    const float* n0 = nodes + 7 * (size_t)i;
    const float* n1 = nodes + 7 * (size_t)(i + 1);
    V3 t0{n0[0], n0[1], n0[2]}; Q4 q0{n0[3], n0[4], n0[5], n0[6]};
    V3 t1{n1[0], n1[1], n1[2]}; Q4 q1{n1[3], n1[4], n1[5], n1[6]};
    V3 v0{vels[3*i+0], vels[3*i+1], vels[3*i+2]};
    V3 v1{vels[3*i+3], vels[3*i+4], vels[3*i+5]};
    V3 dv{imu_dvels[3*i+0], imu_dvels[3*i+1], imu_dvels[3*i+2]};
    V3 dtr{imu_dtrans[3*i+0], imu_dtrans[3*i+1], imu_dtrans[3*i+2]};
    Q4 dr{imu_drots[4*i+0], imu_drots[4*i+1], imu_drots[4*i+2], imu_drots[4*i+3]};
    float dt = dts[i];

    // adjvelerr = imu_dvels - (vels[1:] - vels[:-1]);  L2 = 0.1
    V3 av = v3sub(dv, v3sub(v1, v0));
    out_adjvel[3*i+0] = 0.1f * av.x;
    out_adjvel[3*i+1] = 0.1f * av.y;
    out_adjvel[3*i+2] = 0.1f * av.z;

    // imuroterr = so3_log(qinv(drot) * (qinv(q0) * q1));  L3 = 1.0
    Q4 rq = qmul(qconj(dr), qmul(qconj(q0), q1));
    V3 ir = so3_log(rq);
    out_imurot[3*i+0] = ir.x;
    out_imurot[3*i+1] = ir.y;
    out_imurot[3*i+2] = ir.z;

    // transvelerr = t1 - t0 - (v0*dt + imu_dtrans);  L4 = 0.1
    V3 tv = v3sub(v3sub(t1, t0), v3add(v3scale(dt, v0), dtr));
    out_transvel[3*i+0] = 0.1f * tv.x;
    out_transvel[3*i+1] = 0.1f * tv.y;
    out_transvel[3*i+2] = 0.1f * tv.z;
  }

  if (blockIdx.x == 0 && threadIdx.x < 9) {
    const int stops[3] = {0, 1000, 50000};
    const int j = (int)threadIdx.x;
    // L2 * 100 = 10.0
    out_stop[j] = 10.f * vels[3 * stops[j / 3] + (j % 3)];
  }
}

extern "C" void kernel_launch(void* const* d_in, const int* in_sizes, int n_in,
                              void* d_out, int out_size, void* d_ws, size_t ws_size,
                              hipStream_t stream)
{
  const float* nodes      = (const float*)d_in[0];
  const float* vels       = (const float*)d_in[1];
  const float* poses      = (const float*)d_in[2];
  const float* imu_drots  = (const float*)d_in[3];
  const float* imu_dvels  = (const float*)d_in[4];
  const float* imu_dtrans = (const float*)d_in[5];
  const float* dts        = (const float*)d_in[6];
  const int*   edges      = (const int*)d_in[7];

  const int N = in_sizes[1] / 3;
  const int E = in_sizes[7] / 2;
  const int M = N - 1;

  float* out          = (float*)d_out;
  float* out_adjvel   = out + (size_t)6 * E;
  float* out_imurot   = out_adjvel + (size_t)3 * M;
  float* out_transvel = out_imurot + (size_t)3 * M;
  float* out_stop     = out_transvel + (size_t)3 * M;

  const int gridE = (E + 255) / 256;
  pg_edge_kernel<<<gridE, 256, 0, stream>>>(nodes, poses, edges, out, E);

  const int gridM = (M + 255) / 256;
  pg_adj_kernel<<<gridM, 256, 0, stream>>>(nodes, vels, imu_drots, imu_dvels,
                                           imu_dtrans, dts,
                                           out_adjvel, out_imurot, out_transvel,
                                           out_stop, M);
}